// IntraGraphAttention_4501125726640
// MI455X (gfx1250) — compile-verified
//
#include <hip/hip_runtime.h>

#define NN 100000      // nodes
#define NE 3200000     // edges
#define DI 256         // input dim
#define DO 64          // HEADS*D_HEAD
#define DH 32          // head dim
#define NEG_SLOPE 0.2f

typedef __attribute__((ext_vector_type(2))) float v2f;
typedef __attribute__((ext_vector_type(8))) float v8f;

__device__ __forceinline__ float elu_f(float x) {
    return x > 0.f ? x : (__expf(x) - 1.f);
}
// order-preserving bijection float -> uint so integer atomicMax == float max
__device__ __forceinline__ unsigned f2mono(float f) {
    unsigned u = __float_as_uint(f);
    return (u & 0x80000000u) ? ~u : (u | 0x80000000u);
}
__device__ __forceinline__ float mono2f(unsigned u) {
    return __uint_as_float((u & 0x80000000u) ? (u ^ 0x80000000u) : ~u);
}

// ---------------------------------------------------------------------------
// Kernel 1: h = elu(x) @ W   via V_WMMA_F32_16X16X4_F32
// One wave computes a 16-row x 64-col stripe (4 col-tiles, shared A fragment).
// W staged in LDS in a K-PAIRED layout: lwp[(k>>1)*128 + col*2 + (k&1)], so a
// B fragment {W[kb][col], W[kb+1][col]} (kb even) is one aligned ds_load_b64
// directly into a v2f -- no register re-packing before the WMMA.
// A stream software-pipelined: prefetch next 16-K slab (4 x v2f) while
// running the current slab's 16 WMMAs.
// ---------------------------------------------------------------------------
__global__ void k_gemm_elu_wmma(const float* __restrict__ x,
                                const float* __restrict__ w,
                                float* __restrict__ h) {
    __shared__ float lwp[DI * DO];              // 64 KB, paired layout
    const int tid = threadIdx.x;
    for (int i = tid; i < DI * DO; i += 256) {
        const int k = i / DO, col = i % DO;     // w is [DI][DO] row-major
        lwp[(k >> 1) * (DO * 2) + col * 2 + (k & 1)] = w[i];
    }
    __syncthreads();

    const int wave = tid >> 5;
    const int lane = tid & 31;
    const int tile = blockIdx.x * 8 + wave;     // 16-row tile index
    if (tile * 16 >= NN) return;                // wave-uniform exit (after sync)
    const int r0   = tile * 16;
    const int hi   = lane >> 4;                 // 0: lanes 0-15, 1: lanes 16-31
    const int lsub = lane & 15;
    const int arow = r0 + lsub;                 // A-matrix M = lane%16

    v8f acc0 = {}, acc1 = {}, acc2 = {}, acc3 = {};

    // per-lane A base: row arow, starting float offset hi*2 (8B aligned)
    const float* xbase = x + (size_t)arow * DI + hi * 2;

    v2f araw[4];
#pragma unroll
    for (int j = 0; j < 4; ++j)
        araw[j] = *(const v2f*)(xbase + j * 4);     // slab k=0..15

    for (int k = 0; k < DI; k += 16) {
        v2f anext[4] = {};
        if (k + 16 < DI) {
#pragma unroll
            for (int j = 0; j < 4; ++j)
                anext[j] = *(const v2f*)(xbase + (k + 16) + j * 4);
        }
#pragma unroll
        for (int j = 0; j < 4; ++j) {
            v2f a;
            a.x = elu_f(araw[j].x);
            a.y = elu_f(araw[j].y);
            const int kb = k + j * 4 + hi * 2;      // even: B rows K{kb,kb+1}
            const v2f* bp = (const v2f*)(lwp + (kb >> 1) * (DO * 2) + lsub * 2);
            {
                v2f b = bp[0];
                acc0 = __builtin_amdgcn_wmma_f32_16x16x4_f32(false, a, false, b, (short)0, acc0, false, false);
            }
            {
                v2f b = bp[16];
                acc1 = __builtin_amdgcn_wmma_f32_16x16x4_f32(false, a, false, b, (short)0, acc1, false, false);
            }
            {
                v2f b = bp[32];
                acc2 = __builtin_amdgcn_wmma_f32_16x16x4_f32(false, a, false, b, (short)0, acc2, false, false);
            }
            {
                v2f b = bp[48];
                acc3 = __builtin_amdgcn_wmma_f32_16x16x4_f32(false, a, false, b, (short)0, acc3, false, false);
            }
        }
#pragma unroll
        for (int j = 0; j < 4; ++j) araw[j] = anext[j];
    }

    // D layout: VGPR j -> (M = j + hi*8, N = lane%16)
#pragma unroll
    for (int j = 0; j < 8; ++j) {
        const size_t row = (size_t)(r0 + j + hi * 8);
        float* hr = h + row * DO + lsub;
        hr[ 0] = acc0[j];
        hr[16] = acc1[j];
        hr[32] = acc2[j];
        hr[48] = acc3[j];
    }
}

// ---------------------------------------------------------------------------
// Kernel 2: per-node attention coefficients a_src/a_dst + init accumulators
// ---------------------------------------------------------------------------
__global__ void k_node(const float* __restrict__ h,
                       const float* __restrict__ att_src,
                       const float* __restrict__ att_dst,
                       float* __restrict__ as, float* __restrict__ ad,
                       unsigned* __restrict__ emax, float* __restrict__ denom,
                       float* __restrict__ nw, float* __restrict__ nc,
                       float* __restrict__ out) {
    const int n = blockIdx.x * blockDim.x + threadIdx.x;
    if (n >= NN) return;
    const float* hr = h + (size_t)n * DO;
    float s0 = 0.f, s1 = 0.f, d0 = 0.f, d1 = 0.f;
#pragma unroll
    for (int c = 0; c < DH; ++c) {
        s0 += hr[c]      * att_src[c];
        d0 += hr[c]      * att_dst[c];
        s1 += hr[DH + c] * att_src[DH + c];
        d1 += hr[DH + c] * att_dst[DH + c];
    }
    as[n * 2] = s0;  as[n * 2 + 1] = s1;
    ad[n * 2] = d0;  ad[n * 2 + 1] = d1;
    const unsigned minf = f2mono(-INFINITY);
    emax[n * 2] = minf;  emax[n * 2 + 1] = minf;
    denom[n * 2] = 0.f;  denom[n * 2 + 1] = 0.f;
    nw[n] = 0.f;         nc[n] = 0.f;
    float4* o4 = (float4*)(out + (size_t)n * DO);
#pragma unroll
    for (int i = 0; i < 16; ++i) o4[i] = make_float4(0.f, 0.f, 0.f, 0.f);
}

// ---------------------------------------------------------------------------
// Kernel 3: e = leaky_relu(a_src[s]+a_dst[d]); segment max via uint atomicMax.
// Enrichment stats (real edges only) folded in to avoid re-reading ei[0].
// Edges [0,NE) are real; [NE, NE+NN) are self-loops.
// ---------------------------------------------------------------------------
__global__ void k_edge_max(const long long* __restrict__ ei,
                           const float* __restrict__ ew,
                           const float* __restrict__ as, const float* __restrict__ ad,
                           float* __restrict__ ebuf, unsigned* __restrict__ emax,
                           float* __restrict__ nw, float* __restrict__ nc) {
    const int e = blockIdx.x * blockDim.x + threadIdx.x;
    if (e >= NE + NN) return;
    long long s, d;
    if (e < NE) { s = ei[e]; d = ei[NE + e]; }
    else        { s = (long long)(e - NE); d = s; }
    float e0 = as[s * 2]     + ad[d * 2];
    float e1 = as[s * 2 + 1] + ad[d * 2 + 1];
    e0 = e0 > 0.f ? e0 : NEG_SLOPE * e0;
    e1 = e1 > 0.f ? e1 : NEG_SLOPE * e1;
    ebuf[(size_t)e * 2]     = e0;
    ebuf[(size_t)e * 2 + 1] = e1;
    atomicMax(&emax[d * 2],     f2mono(e0));
    atomicMax(&emax[d * 2 + 1], f2mono(e1));
    if (e < NE) {                               // enrichment stats (src only)
        atomicAdd(&nw[s], ew[e]);
        atomicAdd(&nc[s], 1.0f);
    }
}

// ---------------------------------------------------------------------------
// Kernel 4: ex = exp(e - emax[d]); denom[d] += ex
// ---------------------------------------------------------------------------
__global__ void k_edge_exp(const long long* __restrict__ ei,
                           const unsigned* __restrict__ emax,
                           float* __restrict__ ebuf, float* __restrict__ denom) {
    const int e = blockIdx.x * blockDim.x + threadIdx.x;
    if (e >= NE + NN) return;
    const long long d = (e < NE) ? ei[NE + e] : (long long)(e - NE);
    const float m0 = mono2f(emax[d * 2]);
    const float m1 = mono2f(emax[d * 2 + 1]);
    const float x0 = __expf(ebuf[(size_t)e * 2]     - m0);
    const float x1 = __expf(ebuf[(size_t)e * 2 + 1] - m1);
    ebuf[(size_t)e * 2]     = x0;
    ebuf[(size_t)e * 2 + 1] = x1;
    atomicAdd(&denom[d * 2],     x0);
    atomicAdd(&denom[d * 2 + 1], x1);
}

// ---------------------------------------------------------------------------
// Kernel 5: out[d] += h[s] * alpha  — one wave per edge, lane = channel.
// Lane c covers channels c (head0) and c+32 (head1): coalesced gather +
// 64 lane-contiguous f32 atomics per edge (spread across L2 channels).
// ---------------------------------------------------------------------------
__global__ void k_agg(const long long* __restrict__ ei,
                      const float* __restrict__ h,
                      const float* __restrict__ ebuf,
                      const float* __restrict__ denom,
                      float* __restrict__ out) {
    const int lane   = threadIdx.x & 31;
    const int wid    = blockIdx.x * (blockDim.x >> 5) + (threadIdx.x >> 5);
    const int nwaves = gridDim.x * (blockDim.x >> 5);
    for (int e = wid; e < NE + NN; e += nwaves) {
        long long s, d;
        if (e < NE) { s = ei[e]; d = ei[NE + e]; }
        else        { s = (long long)(e - NE); d = s; }
        const float a0 = ebuf[(size_t)e * 2]     / denom[d * 2];
        const float a1 = ebuf[(size_t)e * 2 + 1] / denom[d * 2 + 1];
        const float* hs = h   + (size_t)s * DO;
        float*       od = out + (size_t)d * DO;
        atomicAdd(&od[lane],      hs[lane]      * a0);
        atomicAdd(&od[lane + 32], hs[lane + 32] * a1);
    }
}

// ---------------------------------------------------------------------------
// Kernel 6: out = (out + bias) * (1 + 0.3*tanh(es)*(clip(nw/max(nc,1)) - 1))
// One wave per node (grid-stride), lane-contiguous access.
// ---------------------------------------------------------------------------
__global__ void k_final(const float* __restrict__ bias,
                        const float* __restrict__ es,
                        const float* __restrict__ nw, const float* __restrict__ nc,
                        float* __restrict__ out) {
    const int lane   = threadIdx.x & 31;
    const int wid    = blockIdx.x * (blockDim.x >> 5) + (threadIdx.x >> 5);
    const int nwaves = gridDim.x * (blockDim.x >> 5);
    const float t = 0.3f * tanhf(es[0]);
    for (int n = wid; n < NN; n += nwaves) {
        float avg = nw[n] / fmaxf(nc[n], 1.0f);
        avg = fminf(fmaxf(avg, 0.3f), 3.0f);
        const float factor = 1.0f + t * (avg - 1.0f);
        float* o = out + (size_t)n * DO;
        o[lane]      = (o[lane]      + bias[lane])      * factor;
        o[lane + 32] = (o[lane + 32] + bias[lane + 32]) * factor;
    }
}

// ---------------------------------------------------------------------------
extern "C" void kernel_launch(void* const* d_in, const int* in_sizes, int n_in,
                              void* d_out, int out_size, void* d_ws, size_t ws_size,
                              hipStream_t stream) {
    (void)in_sizes; (void)n_in; (void)out_size; (void)ws_size;
    const float*     x    = (const float*)d_in[0];
    const long long* ei   = (const long long*)d_in[1];   // int64 [2, NE]
    const float*     ew   = (const float*)d_in[2];
    const float*     lw   = (const float*)d_in[3];
    const float*     asr  = (const float*)d_in[4];
    const float*     adr  = (const float*)d_in[5];
    const float*     bias = (const float*)d_in[6];
    const float*     es   = (const float*)d_in[7];
    float*           out  = (float*)d_out;

    char* ws = (char*)d_ws;
    size_t off = 0;
    auto carve = [&](size_t bytes) -> void* {
        void* p = ws + off;
        off += (bytes + 255) & ~(size_t)255;
        return p;
    };
    float*    h     = (float*)   carve((size_t)NN * DO * sizeof(float));      // 25.6 MB
    float*    a_src = (float*)   carve((size_t)NN * 2 * sizeof(float));
    float*    a_dst = (float*)   carve((size_t)NN * 2 * sizeof(float));
    unsigned* emax  = (unsigned*)carve((size_t)NN * 2 * sizeof(unsigned));
    float*    denom = (float*)   carve((size_t)NN * 2 * sizeof(float));
    float*    nw    = (float*)   carve((size_t)NN * sizeof(float));
    float*    nc    = (float*)   carve((size_t)NN * sizeof(float));
    float*    ebuf  = (float*)   carve((size_t)(NE + NN) * 2 * sizeof(float)); // 26.4 MB

    const int EN = NE + NN;
    k_gemm_elu_wmma<<<(NN / 16 + 7) / 8, 256, 0, stream>>>(x, lw, h);
    k_node<<<(NN + 255) / 256, 256, 0, stream>>>(h, asr, adr, a_src, a_dst,
                                                 emax, denom, nw, nc, out);
    k_edge_max<<<(EN + 255) / 256, 256, 0, stream>>>(ei, ew, a_src, a_dst,
                                                     ebuf, emax, nw, nc);
    k_edge_exp<<<(EN + 255) / 256, 256, 0, stream>>>(ei, emax, ebuf, denom);
    k_agg<<<4096, 256, 0, stream>>>(ei, h, ebuf, denom, out);
    k_final<<<2048, 256, 0, stream>>>(bias, es, nw, nc, out);
}